// ROIHead_8589934592514
// MI455X (gfx1250) — compile-verified
//
#include <hip/hip_runtime.h>
#include <hip/hip_bf16.h>
#include <cmath>

typedef float v2f __attribute__((ext_vector_type(2)));
typedef float v8f __attribute__((ext_vector_type(8)));

#define C_CH   512
#define H_F    38
#define W_F    50
#define ROI_S  7
#define N_ROI  128
#define D_IN   (C_CH * ROI_S * ROI_S)   // 25088
#define D_H    4096
#define N_CLASS 21
#define N_LOC   (N_CLASS * 4)           // 84
#define N_HEAD  (N_LOC + N_CLASS)       // 105

// ---------------------------------------------------------------------------
// Kernel 1: adaptive ROI max-pool (torch AdaptiveMaxPool2d semantics).
// One block per ROI; threads stride over (c, i, j) outputs. Negligible cost.
// ---------------------------------------------------------------------------
__global__ void roi_pool_kernel(const float* __restrict__ feats,
                                const float* __restrict__ rois,
                                float* __restrict__ pooled) {
    const int r = blockIdx.x;
    const float inv = 1.0f / 16.0f;
    const int by0 = (int)truncf(rois[r * 4 + 0] * inv);
    const int bx0 = (int)truncf(rois[r * 4 + 1] * inv);
    const int by2 = (int)ceilf (rois[r * 4 + 2] * inv);
    const int bx2 = (int)ceilf (rois[r * 4 + 3] * inv);
    const int Ly = by2 - by0;
    const int Lx = bx2 - bx0;

    for (int idx = threadIdx.x; idx < D_IN; idx += blockDim.x) {
        const int j = idx % ROI_S;
        const int i = (idx / ROI_S) % ROI_S;
        const int c = idx / (ROI_S * ROI_S);
        const int r0 = by0 + (i * Ly) / ROI_S;
        const int r1 = by0 + ((i + 1) * Ly + ROI_S - 1) / ROI_S;
        const int c0 = bx0 + (j * Lx) / ROI_S;
        const int c1 = bx0 + ((j + 1) * Lx + ROI_S - 1) / ROI_S;
        const float* __restrict__ fc = feats + (size_t)c * (H_F * W_F);
        float m = -INFINITY;
        for (int rr = r0; rr < r1; ++rr) {
            const int rcl = min(max(rr, 0), H_F - 1);
            for (int cc = c0; cc < c1; ++cc) {
                const int ccl = min(max(cc, 0), W_F - 1);
                m = fmaxf(m, fc[rcl * W_F + ccl]);
            }
        }
        pooled[(size_t)r * D_IN + idx] = m;
    }
}

// ---------------------------------------------------------------------------
// Kernel 2: f32 WMMA GEMM  C[M,N] = relu?(A[M,K] @ B[K,N] + bias)
// Block = 256 threads (8 wave32). Block computes a 16(M) x 128(N) tile.
// A tile staged via LDS (coalesced float4 loads, +4 pad kills bank conflicts).
// B (the huge weight stream) loaded non-temporally: streamed once from HBM,
// doesn't evict the small activation matrix from L2.
// ---------------------------------------------------------------------------
#define KC 64
#define LDS_STRIDE (KC + 4)

__global__ __launch_bounds__(256) void wmma_gemm_kernel(
    const float* __restrict__ A,
    const float* __restrict__ B,
    const float* __restrict__ bias,
    float* __restrict__ Cout,
    int N, int K, int do_relu)
{
    __shared__ float lds[16 * LDS_STRIDE];

    const int tid   = threadIdx.x;
    const int lane  = tid & 31;
    const int wave  = tid >> 5;
    const int m0    = blockIdx.y * 16;
    const int ncol_w = blockIdx.x * 128 + wave * 16;
    const int mlane = lane & 15;     // within-tile N (for B/C) or M (for A)
    const int half  = lane >> 4;     // K-half selector

    // cooperative A staging: each thread owns one float4 per KC chunk
    const int lrow = tid >> 4;           // 0..15  (M row of tile)
    const int lcol = (tid & 15) * 4;     // 0..60  (K offset)

    const float* __restrict__ Bp = B + (ncol_w + mlane);

    v8f acc = {};

    for (int k0 = 0; k0 < K; k0 += KC) {
        __syncthreads();
        {
            const float4 av =
                *(const float4*)(A + (size_t)(m0 + lrow) * K + (k0 + lcol));
            float* dst = &lds[lrow * LDS_STRIDE + lcol];
            dst[0] = av.x; dst[1] = av.y; dst[2] = av.z; dst[3] = av.w;
        }
        __syncthreads();

#pragma unroll
        for (int kk = 0; kk < KC; kk += 4) {
            // A fragment: A[mlane, k0+kk+2*half+j]  (from LDS, ds_load_b64)
            v2f a;
            const float* ap = &lds[mlane * LDS_STRIDE + kk + half * 2];
            a.x = ap[0];
            a.y = ap[1];
            // B fragment: B[k0+kk+2*half+j, ncol_w+mlane]  (NT global stream)
            v2f b;
            const size_t krow = (size_t)(k0 + kk + half * 2) * (size_t)N;
            b.x = __builtin_nontemporal_load(Bp + krow);
            b.y = __builtin_nontemporal_load(Bp + krow + N);
            acc = __builtin_amdgcn_wmma_f32_16x16x4_f32(
                false, a, false, b, (short)0, acc, false, false);
        }
    }

    // Epilogue: D VGPR r holds C[m0 + 8*half + r, ncol_w + mlane]
    const int ncol = ncol_w + mlane;
    const float bv = bias[ncol];
#pragma unroll
    for (int r = 0; r < 8; ++r) {
        float v = acc[r] + bv;
        if (do_relu) v = fmaxf(v, 0.0f);
        Cout[(size_t)(m0 + half * 8 + r) * N + ncol] = v;
    }
}

// ---------------------------------------------------------------------------
// Kernel 3: tiny output heads (N=84 and N=21, not WMMA-shaped; ~0.1 GFLOP).
// One thread per output element; writes directly into concatenated d_out.
// ---------------------------------------------------------------------------
__global__ void heads_kernel(const float* __restrict__ h,
                             const float* __restrict__ Wloc,
                             const float* __restrict__ bloc,
                             const float* __restrict__ Wsc,
                             const float* __restrict__ bsc,
                             float* __restrict__ out)
{
    const int t = blockIdx.x * blockDim.x + threadIdx.x;
    if (t >= N_ROI * N_HEAD) return;
    const int r = t / N_HEAD;
    const int o = t % N_HEAD;
    const float* __restrict__ hr = h + (size_t)r * D_H;
    if (o < N_LOC) {
        float s = bloc[o];
        for (int k = 0; k < D_H; ++k)
            s += hr[k] * Wloc[(size_t)k * N_LOC + o];
        out[(size_t)r * N_LOC + o] = s;
    } else {
        const int oo = o - N_LOC;
        float s = bsc[oo];
        for (int k = 0; k < D_H; ++k)
            s += hr[k] * Wsc[(size_t)k * N_CLASS + oo];
        out[(size_t)N_ROI * N_LOC + (size_t)r * N_CLASS + oo] = s;
    }
}

// ---------------------------------------------------------------------------
extern "C" void kernel_launch(void* const* d_in, const int* in_sizes, int n_in,
                              void* d_out, int out_size, void* d_ws, size_t ws_size,
                              hipStream_t stream) {
    const float* features = (const float*)d_in[0];
    const float* rois     = (const float*)d_in[1];
    const float* W1       = (const float*)d_in[2];
    const float* b1       = (const float*)d_in[3];
    const float* W2       = (const float*)d_in[4];
    const float* b2       = (const float*)d_in[5];
    const float* W_loc    = (const float*)d_in[6];
    const float* b_loc    = (const float*)d_in[7];
    const float* W_score  = (const float*)d_in[8];
    const float* b_score  = (const float*)d_in[9];

    float* ws     = (float*)d_ws;
    float* pooled = ws;                              // 128*25088 = 3,211,264 f
    float* h1     = pooled + (size_t)N_ROI * D_IN;   // 128*4096  =   524,288 f
    float* h2     = h1     + (size_t)N_ROI * D_H;    // 128*4096

    roi_pool_kernel<<<N_ROI, 256, 0, stream>>>(features, rois, pooled);

    wmma_gemm_kernel<<<dim3(D_H / 128, N_ROI / 16), 256, 0, stream>>>(
        pooled, W1, b1, h1, D_H, D_IN, 1);

    wmma_gemm_kernel<<<dim3(D_H / 128, N_ROI / 16), 256, 0, stream>>>(
        h1, W2, b2, h2, D_H, D_H, 1);

    heads_kernel<<<(N_ROI * N_HEAD + 255) / 256, 256, 0, stream>>>(
        h2, W_loc, b_loc, W_score, b_score, (float*)d_out);
}